// MambaInMamba_33509334844067
// MI455X (gfx1250) — compile-verified
//
#include <hip/hip_runtime.h>
#include <hip/hip_bf16.h>

typedef float v2f __attribute__((ext_vector_type(2)));
typedef float v8f __attribute__((ext_vector_type(8)));

// ---------------------------------------------------------------------------
// CDNA5 async direct-to-LDS copy (GLOBAL_LOAD_ASYNC_TO_LDS_B128, ASYNCcnt)
// ---------------------------------------------------------------------------
__device__ __forceinline__ void async_copy_b128(const float* gsrc, float* lds_dst)
{
    unsigned l = (unsigned)(uintptr_t)lds_dst;   // generic LDS addr[31:0] = offset
    asm volatile("global_load_async_to_lds_b128 %0, %1, off"
                 :: "v"(l), "v"(gsrc) : "memory");
}

__device__ __forceinline__ void wait_async0()
{
#if __has_builtin(__builtin_amdgcn_s_wait_asynccnt)
    __builtin_amdgcn_s_wait_asynccnt(0);
#else
    asm volatile("s_wait_asynccnt 0x0" ::: "memory");
#endif
}

// ---------------------------------------------------------------------------
// Generic NT GEMM:  C[m,n] = sum_k A[m,k] * Bw[n,k]   (fp32, WMMA 16x16x4)
// Block tile 64x64, 256 threads = 8 waves, wave tile 16x32 (2 WMMAs / k-step).
// Double-buffered LDS fed by async global->LDS copies; one barrier per chunk.
// epilogue: 0 = none, 1 = clamp(C + bias[n], -6, 6)
// ---------------------------------------------------------------------------
#define TM 64
#define TN 64
#define KC 16
#define LDS_STRIDE 20   // padded (20*4B = 5 banks) to spread bank use, 16B aligned
#define TSLOT (TM * LDS_STRIDE)

__global__ __launch_bounds__(256) void gemm_nt_f32_wmma(
    const float* __restrict__ A, int lda,
    const float* __restrict__ Bw, int ldb,
    float* __restrict__ C, int ldc,
    int M, int N, int K,
    const float* __restrict__ bias, int epilogue)
{
    __shared__ float As[2 * TSLOT];
    __shared__ float Bs[2 * TSLOT];

    const int bm   = blockIdx.y * TM;
    const int bn   = blockIdx.x * TN;
    const int tid  = threadIdx.x;
    const int wave = tid >> 5;
    const int lane = tid & 31;
    const int wrow = (wave >> 1) * 16;   // 0,16,32,48
    const int wcol = (wave & 1) * 32;    // 0,32
    const int lhalf = lane >> 4;         // 0/1 -> K pair select
    const int lmod  = lane & 15;

    // cooperative tile-load mapping: one b128 slot of A and of B per thread
    const int lrow = tid >> 2;           // 0..63
    const int lk4  = (tid & 3) * 4;      // 0,4,8,12
    const int ar   = bm + lrow;
    const int br   = bn + lrow;

    v8f acc0 = {};
    v8f acc1 = {};

    const int nk = (K + KC - 1) / KC;
    const float4 f4z = {0.f, 0.f, 0.f, 0.f};

    // issue one K-chunk's tiles into LDS buffer `buf` (async, zero-fill edges)
    auto issue_tiles = [&](int k0, int buf) {
        const int ak = k0 + lk4;
        float* adst = &As[buf * TSLOT + lrow * LDS_STRIDE + lk4];
        float* bdst = &Bs[buf * TSLOT + lrow * LDS_STRIDE + lk4];
        if (ar < M && ak < K)
            async_copy_b128(A + (size_t)ar * lda + ak, adst);
        else
            *(float4*)adst = f4z;
        if (br < N && ak < K)
            async_copy_b128(Bw + (size_t)br * ldb + ak, bdst);
        else
            *(float4*)bdst = f4z;
    };

    issue_tiles(0, 0);   // prologue

    for (int kc = 0; kc < nk; ++kc) {
        const int cur = kc & 1;
        wait_async0();        // own current-tile async copies have landed in LDS
        __syncthreads();      // everyone's landed; everyone done reading `nxt`
        if (kc + 1 < nk)
            issue_tiles((kc + 1) * KC, cur ^ 1);   // overlap with compute below

        const float* cA = &As[cur * TSLOT];
        const float* cB = &Bs[cur * TSLOT];
        #pragma unroll
        for (int kk = 0; kk < KC; kk += 4) {
            // A frag (16x4): lanes 0-15 hold K=kk,kk+1 ; lanes 16-31 K=kk+2,kk+3
            v2f a, b0, b1;
            a[0]  = cA[(wrow + lmod) * LDS_STRIDE + kk + 2 * lhalf];
            a[1]  = cA[(wrow + lmod) * LDS_STRIDE + kk + 2 * lhalf + 1];
            b0[0] = cB[(wcol + lmod) * LDS_STRIDE + kk + 2 * lhalf];
            b0[1] = cB[(wcol + lmod) * LDS_STRIDE + kk + 2 * lhalf + 1];
            b1[0] = cB[(wcol + 16 + lmod) * LDS_STRIDE + kk + 2 * lhalf];
            b1[1] = cB[(wcol + 16 + lmod) * LDS_STRIDE + kk + 2 * lhalf + 1];
            acc0 = __builtin_amdgcn_wmma_f32_16x16x4_f32(
                false, a, false, b0, (short)0, acc0, false, false);
            acc1 = __builtin_amdgcn_wmma_f32_16x16x4_f32(
                false, a, false, b1, (short)0, acc1, false, false);
        }
    }

    // C/D layout: VGPR i -> row = wrow + i + 8*lhalf, col = lmod (+16 for acc1)
    #pragma unroll
    for (int i = 0; i < 8; ++i) {
        const int row  = bm + wrow + i + 8 * lhalf;
        const int col0 = bn + wcol + lmod;
        const int col1 = col0 + 16;
        float v0 = acc0[i];
        float v1 = acc1[i];
        if (row < M) {
            if (col0 < N) {
                if (epilogue == 1)
                    v0 = fminf(fmaxf(v0 + bias[col0], -6.0f), 6.0f);
                C[(size_t)row * ldc + col0] = v0;
            }
            if (col1 < N) {
                if (epilogue == 1)
                    v1 = fminf(fmaxf(v1 + bias[col1], -6.0f), 6.0f);
                C[(size_t)row * ldc + col1] = v1;
            }
        }
    }
}

// ---------------------------------------------------------------------------
// Causal depthwise conv (width W) + SiLU.  in: (b, l, ld_in) taking first D
// channels; out: (b, l, D) contiguous.
// ---------------------------------------------------------------------------
template <int W>
__global__ __launch_bounds__(256) void dwconv_silu(
    const float* __restrict__ in, int ld_in,
    const float* __restrict__ w, const float* __restrict__ b,
    float* __restrict__ out, int D, int L, int NB)
{
    int idx = blockIdx.x * blockDim.x + threadIdx.x;
    int total = NB * L * D;
    if (idx >= total) return;
    int d  = idx % D;
    int t  = (idx / D) % L;
    int bb = idx / (D * L);
    const float* base = in + (size_t)bb * L * ld_in + d;
    float acc = b[d];
    #pragma unroll
    for (int i = 0; i < W; ++i) {
        int tt = t - (W - 1 - i);
        if (tt >= 0) acc += base[(size_t)tt * ld_in] * w[d * W + i];
    }
    float sig = 1.0f / (1.0f + __expf(-acc));
    out[(size_t)(bb * L + t) * D + d] = acc * sig;
}

// ---------------------------------------------------------------------------
// Selective scan. One lane = one channel d; state (NS floats) in registers.
// B/C state vectors (shared across channels of a batch) staged via LDS in
// 32-timestep chunks (broadcast reads). Sequential over L.
//   y[b,t,d] = (sum_j s_j*C[t,j] + D[d]*u) * silu(z),
//   s_j <- s_j*exp(dt*A[d,j]) + dt*u*B[t,j],  dt = softplus(delta + bias[d])
// ---------------------------------------------------------------------------
template <int NS>
__global__ __launch_bounds__(256) void selective_scan(
    const float* __restrict__ u,     int ldu,
    const float* __restrict__ delta, int ldd,
    const float* __restrict__ A_log,           // Dch x NS
    const float* __restrict__ Bm,    int ldB,  // row (b*L+t), NS cols
    const float* __restrict__ Cm,    int ldC,
    const float* __restrict__ Dv,
    const float* __restrict__ zp,    int ldz,
    const float* __restrict__ bias,
    float* __restrict__ yp,          int ldy,
    int Dch, int L)
{
    const int TCH = 32;
    __shared__ float sB[TCH * NS];
    __shared__ float sC[TCH * NS];

    const int d     = blockIdx.x * blockDim.x + threadIdx.x;  // channel
    const int batch = blockIdx.y;
    const float bd  = bias[d];
    const float Dd  = Dv[d];

    float Ar[NS];
    float s[NS];
    #pragma unroll
    for (int j = 0; j < NS; ++j) {
        Ar[j] = -__expf(A_log[(size_t)d * NS + j]);
        s[j]  = 0.0f;
    }

    for (int t0 = 0; t0 < L; t0 += TCH) {
        __syncthreads();
        for (int i = threadIdx.x; i < TCH * NS; i += blockDim.x) {
            int tt = i / NS, j = i % NS;
            size_t row = (size_t)(batch * L + t0 + tt);
            sB[i] = Bm[row * ldB + j];
            sC[i] = Cm[row * ldC + j];
        }
        __syncthreads();

        for (int tt = 0; tt < TCH; ++tt) {
            size_t row = (size_t)(batch * L + t0 + tt);
            float dt = delta[row * ldd + d] + bd;
            dt = (dt > 20.0f) ? dt : __logf(1.0f + __expf(dt));  // softplus
            float ut = u[row * ldu + d];
            float du = dt * ut;
            float acc = 0.0f;
            #pragma unroll
            for (int j = 0; j < NS; ++j) {
                s[j] = s[j] * __expf(dt * Ar[j]) + du * sB[tt * NS + j];
                acc += s[j] * sC[tt * NS + j];
            }
            float zt  = zp[row * ldz + d];
            float sig = 1.0f / (1.0f + __expf(-zt));
            yp[row * ldy + d] = (acc + Dd * ut) * (zt * sig);
        }
    }
}

// ---------------------------------------------------------------------------
// Orchestration
// ---------------------------------------------------------------------------
static inline void launch_gemm(const float* A, int lda, const float* Bw, int ldb,
                               float* C, int ldc, int M, int N, int K,
                               const float* bias, int ep, hipStream_t stream)
{
    dim3 grid((N + TN - 1) / TN, (M + TM - 1) / TM);
    gemm_nt_f32_wmma<<<grid, 256, 0, stream>>>(A, lda, Bw, ldb, C, ldc,
                                               M, N, K, bias, ep);
}

extern "C" void kernel_launch(void* const* d_in, const int* in_sizes, int n_in,
                              void* d_out, int out_size, void* d_ws, size_t ws_size,
                              hipStream_t stream)
{
    (void)in_sizes; (void)n_in; (void)out_size; (void)ws_size;

    // dims
    const int B = 2, L = 1024, DM = 1024, DS = 16, DI = 2048, R = 64;
    const int MDM = 64, MDS = 4, MR = 4;
    const int BL = B * L;  // 2048

    // inputs (setup_inputs order, all fp32)
    const float* h        = (const float*)d_in[0];   // (B,L,DM)
    const float* in_w     = (const float*)d_in[1];   // (2*DI, DM)
    const float* conv_w   = (const float*)d_in[2];   // (DI, 4)
    const float* conv_b   = (const float*)d_in[3];   // (DI)
    const float* xproj_w  = (const float*)d_in[4];   // (R+2*DS, DI) = (96,2048)
    const float* dt_out_w = (const float*)d_in[5];   // (DI, R)
    const float* dt_out_b = (const float*)d_in[6];   // (DI)
    const float* A_log    = (const float*)d_in[7];   // (DI, DS)
    const float* Dv       = (const float*)d_in[8];   // (DI)
    const float* out_w    = (const float*)d_in[9];   // (DM, DI)
    const float* m_in_w   = (const float*)d_in[10];  // (2*MDM, MDM)
    const float* m_conv_w = (const float*)d_in[11];  // (MDM, 2)
    const float* m_conv_b = (const float*)d_in[12];  // (MDM)
    const float* m_xproj  = (const float*)d_in[13];  // (MR+2*MDS, MDM) = (12,64)
    const float* m_dt_w   = (const float*)d_in[14];  // (MDM, MR)
    const float* m_dt_b   = (const float*)d_in[15];  // (MDM)
    const float* m_A_log  = (const float*)d_in[16];  // (MDM, MDS)
    const float* m_Dv     = (const float*)d_in[17];  // (MDM)
    const float* m_out_w  = (const float*)d_in[18];  // (MDM, MDM)

    float* out = (float*)d_out;                       // (B,L,DM)

    // workspace layout (floats)
    float* ws      = (float*)d_ws;
    float* xz      = ws;                              // (BL, 2*DI)   8388608
    float* xconv   = xz      + (size_t)BL * 2 * DI;   // (BL, DI)     4194304
    float* xdbl    = xconv   + (size_t)BL * DI;       // (BL, 96)      196608
    float* xzm     = xdbl    + (size_t)BL * 96;       // (BL, 128)     262144
    float* xmconv  = xzm     + (size_t)BL * 128;      // (BL, 64)      131072
    float* xdblm   = xmconv  + (size_t)BL * 64;       // (BL, 12)       24576
    float* deltam  = xdblm   + (size_t)BL * 12;       // (BL, 64)      131072
    float* ym      = deltam  + (size_t)BL * 64;       // (BL, 64)      131072
    float* dtm     = ym      + (size_t)BL * 64;       // (BL, 64)      131072
    float* delta   = dtm     + (size_t)BL * 64;       // (BL, DI)     4194304
    float* yv      = delta   + (size_t)BL * DI;       // (BL, DI)     4194304

    // 1. xz[(b,l), e] = h @ in_w^T    (M=2048, N=4096, K=1024)
    launch_gemm(h, DM, in_w, DM, xz, 2 * DI, BL, 2 * DI, DM, nullptr, 0, stream);

    // 2. causal conv W=4 + SiLU on x half of xz -> xconv (b,l,DI)
    {
        int total = B * L * DI;
        dwconv_silu<4><<<(total + 255) / 256, 256, 0, stream>>>(
            xz, 2 * DI, conv_w, conv_b, xconv, DI, L, B);
    }

    // 3. xdbl = xconv @ xproj_w^T    (M=2048, N=96, K=2048)
    launch_gemm(xconv, DI, xproj_w, DI, xdbl, 96, BL, 96, DI, nullptr, 0, stream);

    // 4. inner in-proj: xzm = dt_in(xdbl[:, :64]) @ m_in_w^T  (N=128, K=64)
    launch_gemm(xdbl, 96, m_in_w, MDM, xzm, 2 * MDM, BL, 2 * MDM, R, nullptr, 0, stream);

    // 5. inner conv W=2 + SiLU -> xmconv (b,l,64)
    {
        int total = B * L * MDM;
        dwconv_silu<2><<<(total + 255) / 256, 256, 0, stream>>>(
            xzm, 2 * MDM, m_conv_w, m_conv_b, xmconv, MDM, L, B);
    }

    // 6. xdblm = xmconv @ m_xproj^T   (N=12, K=64)
    launch_gemm(xmconv, MDM, m_xproj, MDM, xdblm, 12, BL, 12, MDM, nullptr, 0, stream);

    // 7. deltam = xdblm[:, :4] @ m_dt_w^T   (N=64, K=4)
    launch_gemm(xdblm, 12, m_dt_w, MR, deltam, MDM, BL, MDM, MR, nullptr, 0, stream);

    // 8. inner selective scan (state = 4) -> ym (b,l,64)
    {
        dim3 grid(MDM / 64, B);
        selective_scan<MDS><<<grid, 64, 0, stream>>>(
            xmconv, MDM, deltam, MDM, m_A_log,
            xdblm + MR, 12, xdblm + MR + MDS, 12,
            m_Dv, xzm + MDM, 2 * MDM, m_dt_b, ym, MDM, MDM, L);
    }

    // 9. dtm = ym @ m_out_w^T   (N=64, K=64)
    launch_gemm(ym, MDM, m_out_w, MDM, dtm, MDM, BL, MDM, MDM, nullptr, 0, stream);

    // 10. delta = clamp(dtm @ dt_out_w^T + dt_out_b, -6, 6)  (N=2048, K=64)
    launch_gemm(dtm, MDM, dt_out_w, R, delta, DI, BL, DI, R, dt_out_b, 1, stream);

    // 11. main selective scan (state = 16) -> yv (b,l,DI)
    {
        dim3 grid(DI / 256, B);
        selective_scan<DS><<<grid, 256, 0, stream>>>(
            xconv, DI, delta, DI, A_log,
            xdbl + R, 96, xdbl + R + DS, 96,
            Dv, xz + DI, 2 * DI, dt_out_b, yv, DI, DI, L);
    }

    // 12. out = yv @ out_w^T   (M=2048, N=1024, K=2048)
    launch_gemm(yv, DI, out_w, DI, out, DM, BL, DM, DI, nullptr, 0, stream);
}